// FactorizedVectorQuantize_26439818674710
// MI455X (gfx1250) — compile-verified
//
#include <hip/hip_runtime.h>
#include <stdint.h>

typedef __attribute__((ext_vector_type(2))) float v2f;
typedef __attribute__((ext_vector_type(8))) float v8f;

#define Bx   16
#define Dx   1024
#define Tx   4096
#define CSx  1024
#define CDx  8
#define TT   256      // t-tile per block
#define NT   256      // threads per block (8 waves of 32)
#define EPSF 1e-12f

__device__ __forceinline__ float warp_sum(float v) {
  #pragma unroll
  for (int off = 16; off > 0; off >>= 1) v += __shfl_xor(v, off, 32);
  return v;
}

// Async global->LDS copy, 16 bytes per lane (GLOBAL_LOAD_ASYNC_TO_LDS_B128,
// GVS mode: mem addr = SGPR base + per-lane 32-bit byte offset). Tracked by
// ASYNCcnt; completion enforced with s_wait_asynccnt before the barrier.
__device__ __forceinline__ void async_ld_b128(void* lds_ptr, const void* gbase,
                                              uint32_t byte_off) {
  uint32_t lds_addr = (uint32_t)(uintptr_t)lds_ptr;  // low 32 bits = LDS offset
  asm volatile("global_load_async_to_lds_b128 %0, %1, %2"
               :: "v"(lds_addr), "v"(byte_off), "s"(gbase)
               : "memory");
}

__device__ __forceinline__ void wait_async_zero() {
#if __has_builtin(__builtin_amdgcn_s_wait_asynccnt)
  __builtin_amdgcn_s_wait_asynccnt(0);
#else
  asm volatile("s_wait_asynccnt 0x0" ::: "memory");
#endif
}

// ---------------------------------------------------------------------------
// Prep: weight_norm(W_in), weight_norm(W_out), L2-normalized codebook -> ws
// ---------------------------------------------------------------------------
__global__ __launch_bounds__(NT)
void fvq_prep(const float* __restrict__ cb,
              const float* __restrict__ g_in,  const float* __restrict__ v_in,
              const float* __restrict__ g_out, const float* __restrict__ v_out,
              float* __restrict__ Win, float* __restrict__ Wout,
              float* __restrict__ Cbn) {
  const int tid  = threadIdx.x;
  const int lane = tid & 31;
  const int wave = tid >> 5;

  // W_in: 8 rows of 1024, one wave per row (dim=0 weight norm)
  {
    const int c = wave;
    float s = 0.f;
    for (int j = lane; j < Dx; j += 32) { float x = v_in[c * Dx + j]; s += x * x; }
    s = warp_sum(s);
    const float scale = g_in[c] / fmaxf(sqrtf(s), EPSF);
    for (int j = lane; j < Dx; j += 32) Win[c * Dx + j] = v_in[c * Dx + j] * scale;
  }
  // W_out: 1024 rows of 8, one thread per row
  for (int r = tid; r < Dx; r += NT) {
    float s = 0.f;
    #pragma unroll
    for (int c = 0; c < CDx; c++) { float x = v_out[r * CDx + c]; s += x * x; }
    const float scale = g_out[r] / fmaxf(sqrtf(s), EPSF);
    #pragma unroll
    for (int c = 0; c < CDx; c++) Wout[r * CDx + c] = v_out[r * CDx + c] * scale;
  }
  // Normalized codebook: 1024 rows of 8
  for (int r = tid; r < CSx; r += NT) {
    float s = 0.f;
    #pragma unroll
    for (int c = 0; c < CDx; c++) { float x = cb[r * CDx + c]; s += x * x; }
    const float inv = 1.0f / fmaxf(sqrtf(s), EPSF);
    #pragma unroll
    for (int c = 0; c < CDx; c++) Cbn[r * CDx + c] = cb[r * CDx + c] * inv;
  }
}

// ---------------------------------------------------------------------------
// Fused: in_proj -> WMMA cosine-NN argmax -> gather/loss -> out_proj
// grid = (Tx/TT, Bx), block = NT threads
// ---------------------------------------------------------------------------
__global__ __launch_bounds__(NT)
void fvq_main(const float* __restrict__ z,    const float* __restrict__ cb,
              const float* __restrict__ b_in, const float* __restrict__ b_out,
              const float* __restrict__ Win,  const float* __restrict__ Wout,
              const float* __restrict__ Cbn,
              float* __restrict__ out, float* __restrict__ outIdx,
              float* __restrict__ lossPart) {
  __shared__ float sWin[CDx][Dx];    // 32 KB
  __shared__ float sWout[Dx][CDx];   // 32 KB
  __shared__ float sBout[Dx];        //  4 KB
  __shared__ float sCb[CSx][CDx];    // 32 KB (normalized codebook)
  __shared__ float sZe[TT][CDx];     //  8 KB (raw z_e for loss)
  __shared__ float sEn[TT][CDx];     //  8 KB (normalized z_e)
  __shared__ int   sIdx[TT];         //  1 KB
  __shared__ float sRed[NT / 32];

  const int tid  = threadIdx.x;
  const int b    = blockIdx.y;
  const int t0   = blockIdx.x * TT;
  const int t    = t0 + tid;
  const int lane = tid & 31;
  const int wave = tid >> 5;

  // ---- stage small operands into LDS with async global->LDS b128 copies ----
  {
    float4* dWin  = (float4*)&sWin[0][0];
    float4* dWout = (float4*)&sWout[0][0];
    float4* dCb   = (float4*)&sCb[0][0];
    float4* dBout = (float4*)sBout;
    for (int i = tid; i < CDx * Dx / 4; i += NT)
      async_ld_b128(&dWin[i], Win, (uint32_t)(i * 16));
    for (int i = tid; i < Dx * CDx / 4; i += NT)
      async_ld_b128(&dWout[i], Wout, (uint32_t)(i * 16));
    for (int i = tid; i < CSx * CDx / 4; i += NT)
      async_ld_b128(&dCb[i], Cbn, (uint32_t)(i * 16));
    for (int i = tid; i < Dx / 4; i += NT)
      async_ld_b128(&dBout[i], b_out, (uint32_t)(i * 16));
  }
  wait_async_zero();
  __syncthreads();

  // ---- Phase 1: z_e[c] = b_in[c] + sum_d W_in[c,d] * z[b,d,t] ----
  // z is streamed exactly once across the grid -> non-temporal loads.
  float acc[CDx];
  #pragma unroll
  for (int c = 0; c < CDx; c++) acc[c] = b_in[c];
  const float* zp = z + ((size_t)b * Dx) * Tx + t;
  #pragma unroll 8
  for (int d = 0; d < Dx; d++) {
    const float zv = __builtin_nontemporal_load(zp + (size_t)d * Tx);
    #pragma unroll
    for (int c = 0; c < CDx; c++) acc[c] += sWin[c][d] * zv;
  }
  float nrm = 0.f;
  #pragma unroll
  for (int c = 0; c < CDx; c++) { sZe[tid][c] = acc[c]; nrm += acc[c] * acc[c]; }
  const float inv = 1.0f / fmaxf(sqrtf(nrm), EPSF);
  #pragma unroll
  for (int c = 0; c < CDx; c++) sEn[tid][c] = acc[c] * inv;
  __syncthreads();

  // ---- Phase 2: WMMA NN search: dot = enc_n[16t x 8] * cb_n^T[8 x 16codes] ----
  // V_WMMA_F32_16X16X4_F32 operand layout (wave32):
  //   A 16x4: lanes 0-15 hold M=lane, {K0,K1}; lanes 16-31 hold {K2,K3}
  //   B 4x16: lanes 0-15 hold N=lane, {K0,K1}; lanes 16-31 hold {K2,K3}
  //   C/D:    VGPR v: lanes 0-15 -> M=v, N=lane; lanes 16-31 -> M=v+8, N=lane-16
  const int m  = lane & 15;
  const int hi = lane >> 4;
  const int kb = hi * 2;

  for (int tg = wave * 2; tg < wave * 2 + 2; tg++) {
    const int trow = tg * 16 + m;
    v2f a0 = { sEn[trow][kb],     sEn[trow][kb + 1] };       // K-step 0 (K=0..3)
    v2f a1 = { sEn[trow][4 + kb], sEn[trow][4 + kb + 1] };   // K-step 1 (K=4..7)

    float bv[8]; int bi[8];
    #pragma unroll
    for (int v = 0; v < 8; v++) { bv[v] = -3.0e38f; bi[v] = 0; }

    for (int cg = 0; cg < CSx / 16; cg++) {
      const int code = cg * 16 + m;
      v2f b0 = { sCb[code][kb],     sCb[code][kb + 1] };
      v2f b1 = { sCb[code][4 + kb], sCb[code][4 + kb + 1] };
      v8f cacc = {0.f, 0.f, 0.f, 0.f, 0.f, 0.f, 0.f, 0.f};
      cacc = __builtin_amdgcn_wmma_f32_16x16x4_f32(false, a0, false, b0,
                                                   (short)0, cacc, false, false);
      cacc = __builtin_amdgcn_wmma_f32_16x16x4_f32(false, a1, false, b1,
                                                   (short)0, cacc, false, false);
      // per-lane running argmax; code index is monotone in cg, so strict '>'
      // reproduces jnp.argmax first-max tie-breaking within a lane
      #pragma unroll
      for (int v = 0; v < 8; v++) {
        const float val = cacc[v];
        const bool better = val > bv[v];
        bv[v] = better ? val : bv[v];
        bi[v] = better ? code : bi[v];
      }
    }
    // reduce across the 16 lanes that share each t-row (tie -> lowest index)
    #pragma unroll
    for (int v = 0; v < 8; v++) {
      #pragma unroll
      for (int off = 1; off < 16; off <<= 1) {
        const float ov = __shfl_xor(bv[v], off, 32);
        const int   oi = __shfl_xor(bi[v], off, 32);
        const bool better = (ov > bv[v]) || ((ov == bv[v]) && (oi < bi[v]));
        bv[v] = better ? ov : bv[v];
        bi[v] = better ? oi : bi[v];
      }
    }
    if (m == 0) {
      #pragma unroll
      for (int v = 0; v < 8; v++) {
        const int row = tg * 16 + hi * 8 + v;
        sIdx[row] = bi[v];
        outIdx[(size_t)b * Tx + t0 + row] = (float)bi[v];
      }
    }
  }
  __syncthreads();

  // ---- Phase 3: gather raw codebook row, per-block commit-loss partial ----
  const int myIdx = sIdx[tid];
  float zq[CDx];
  {
    const float4* crow = (const float4*)(cb + (size_t)myIdx * CDx);
    const float4 q0 = crow[0], q1 = crow[1];
    zq[0] = q0.x; zq[1] = q0.y; zq[2] = q0.z; zq[3] = q0.w;
    zq[4] = q1.x; zq[5] = q1.y; zq[6] = q1.z; zq[7] = q1.w;
  }
  float l = 0.f;
  #pragma unroll
  for (int c = 0; c < CDx; c++) { const float dlt = sZe[tid][c] - zq[c]; l += dlt * dlt; }
  l = warp_sum(l);
  if (lane == 0) sRed[wave] = l;
  __syncthreads();
  if (tid == 0) {
    float s = 0.f;
    #pragma unroll
    for (int w = 0; w < NT / 32; w++) s += sRed[w];
    lossPart[b * gridDim.x + blockIdx.x] = s;   // deterministic (no atomics)
  }

  // ---- Phase 4: out[b,d,t] = b_out[d] + sum_c W_out[d,c] * z_q[c] ----
  // out is written exactly once -> non-temporal stores.
  float* op = out + ((size_t)b * Dx) * Tx + t;
  for (int d = 0; d < Dx; d++) {
    float o = sBout[d];
    #pragma unroll
    for (int c = 0; c < CDx; c++) o += sWout[d][c] * zq[c];
    __builtin_nontemporal_store(o, op + (size_t)d * Tx);
  }
}

// ---------------------------------------------------------------------------
// Finish: reduce loss partials -> commit_loss[b] = 1.25 * mean((z_e - z_q)^2)
// ---------------------------------------------------------------------------
__global__ void fvq_finish(const float* __restrict__ lossPart,
                           float* __restrict__ lossOut, int parts) {
  const int b = threadIdx.x;
  if (b < Bx) {
    float s = 0.f;
    for (int i = 0; i < parts; i++) s += lossPart[b * parts + i];
    lossOut[b] = 1.25f * s / (float)(CDx * Tx);
  }
}

extern "C" void kernel_launch(void* const* d_in, const int* in_sizes, int n_in,
                              void* d_out, int out_size, void* d_ws, size_t ws_size,
                              hipStream_t stream) {
  (void)in_sizes; (void)n_in; (void)out_size; (void)ws_size;
  const float* z     = (const float*)d_in[0];
  const float* cb    = (const float*)d_in[1];
  const float* g_in  = (const float*)d_in[2];
  const float* v_in  = (const float*)d_in[3];
  const float* b_in  = (const float*)d_in[4];
  const float* g_out = (const float*)d_in[5];
  const float* v_out = (const float*)d_in[6];
  const float* b_out = (const float*)d_in[7];

  float* ws       = (float*)d_ws;
  float* Win      = ws;                    // 8192 floats
  float* Wout     = Win  + CDx * Dx;       // 8192
  float* Cbn      = Wout + Dx * CDx;       // 8192
  float* lossPart = Cbn  + CSx * CDx;      // Bx * (Tx/TT) = 256

  float* outMain = (float*)d_out;                       // [B, D, T]
  float* outIdx  = outMain + (size_t)Bx * Dx * Tx;      // [B, T]
  float* lossOut = outIdx  + (size_t)Bx * Tx;           // [B]

  fvq_prep<<<1, NT, 0, stream>>>(cb, g_in, v_in, g_out, v_out, Win, Wout, Cbn);
  dim3 grid(Tx / TT, Bx);
  fvq_main<<<grid, NT, 0, stream>>>(z, cb, b_in, b_out, Win, Wout, Cbn,
                                    outMain, outIdx, lossPart);
  fvq_finish<<<1, 32, 0, stream>>>(lossPart, lossOut, Tx / TT);
}